// DeformConv2d_79671643340828
// MI455X (gfx1250) — compile-verified
//
#include <hip/hip_runtime.h>

typedef __attribute__((ext_vector_type(2))) float v2f;
typedef __attribute__((ext_vector_type(8))) float v8f;

#define H_    128
#define W_    128
#define HW_   (H_ * W_)      // 16384 = 2^14
#define CIN   64
#define COUT  64
#define NB    4
#define KTOT  576            // 64 ch * 9 taps
#define NCHUNK (KTOT / 4)    // 144 WMMA K-chunks

// ---------------------------------------------------------------------------
// Kernel 0: repack conv_w [64][576] into WMMA A-fragment order.
// Layout: Aswz[(wave*NCHUNK + kc)*32 + lane] = {A[co][k], A[co][k+1]}
//   co = wave*16 + (lane&15),  k = kc*4 + (lane>>4)*2
// so each wave's per-chunk A load is 32 lanes x 8B fully coalesced.
// ---------------------------------------------------------------------------
__global__ void swizzle_A_kernel(const float* __restrict__ conv_w,
                                 v2f* __restrict__ Aswz) {
  int idx = blockIdx.x * blockDim.x + threadIdx.x;
  if (idx >= 4 * NCHUNK * 32) return;
  int lane = idx & 31;
  int kc   = (idx >> 5) % NCHUNK;
  int wv   = (idx >> 5) / NCHUNK;
  int co   = wv * 16 + (lane & 15);
  int k    = (kc << 2) + ((lane >> 4) << 1);
  v2f a;
  a.x = conv_w[co * KTOT + k];
  a.y = conv_w[co * KTOT + k + 1];
  Aswz[idx] = a;
}

// ---------------------------------------------------------------------------
// Kernel 1: offset-predictor conv. 3x3, 64 -> 18 channels, pad 1.
// One thread per output pixel, all 18 channels; p_w staged in LDS
// (uniform per-wave reads -> broadcast, no bank conflicts).
// ---------------------------------------------------------------------------
__global__ __launch_bounds__(256) void offset_conv_kernel(
    const float* __restrict__ x,
    const float* __restrict__ p_w,
    const float* __restrict__ p_b,
    float* __restrict__ off) {
  __shared__ float wlds[18 * KTOT];  // 41.5 KB
  for (int i = threadIdx.x; i < 18 * KTOT; i += blockDim.x) wlds[i] = p_w[i];
  __syncthreads();

  int gid = blockIdx.x * blockDim.x + threadIdx.x;  // 0 .. 4*16384-1
  int b = gid >> 14;
  int p = gid & (HW_ - 1);
  int h = p >> 7;
  int w = p & (W_ - 1);

  float acc[18];
#pragma unroll
  for (int oc = 0; oc < 18; ++oc) acc[oc] = p_b[oc];

  const float* xb = x + (size_t)b * CIN * HW_;
  for (int ci = 0; ci < CIN; ++ci) {
    const float* xc = xb + ci * HW_;
#pragma unroll
    for (int kh = 0; kh < 3; ++kh) {
      int hh = h + kh - 1;
#pragma unroll
      for (int kw = 0; kw < 3; ++kw) {
        int ww = w + kw - 1;
        float xv = (hh >= 0 && hh < H_ && ww >= 0 && ww < W_)
                       ? xc[hh * W_ + ww] : 0.f;
        int kidx = ci * 9 + kh * 3 + kw;
#pragma unroll
        for (int oc = 0; oc < 18; ++oc)
          acc[oc] = fmaf(wlds[oc * KTOT + kidx], xv, acc[oc]);
      }
    }
  }
#pragma unroll
  for (int oc = 0; oc < 18; ++oc)
    off[((size_t)b * 18 + oc) * HW_ + p] = acc[oc];
}

// ---------------------------------------------------------------------------
// Kernel 2: fused bilinear sampling + 64x576 GEMM via V_WMMA_F32_16X16X4_F32.
// One workgroup (4 waves) per 16-pixel tile; wave w computes the
// 16(co) x 16(px) output tile for co in [16w, 16w+16).
// ---------------------------------------------------------------------------
__global__ __launch_bounds__(128) void deform_wmma_kernel(
    const float* __restrict__ x,
    const float* __restrict__ off,
    const v2f* __restrict__ Aswz,
    float* __restrict__ out) {
  __shared__ float Bmat[KTOT * 16];  // 36 KB: Bmat[(ci*9+n)*16 + px]

  int tile = blockIdx.x;        // 4096 tiles: 1024 per batch
  int b    = tile >> 10;
  int p0   = (tile & 1023) << 4;  // 16 consecutive pixels, same row h
  int h    = p0 >> 7;
  int tid  = threadIdx.x;
  int lane = tid & 31;
  int wave = tid >> 5;

  const float* xb   = x   + (size_t)b * CIN * HW_;
  const float* offb = off + (size_t)b * 18 * HW_;

  // ---- Phase 1: build patch matrix in LDS (coords computed once per
  //      (pixel,tap) pair, reused across all 64 channels) -----------------
  for (int pair = tid; pair < 144; pair += 128) {
    int px = pair & 15;
    int n  = pair >> 4;
    int p  = p0 + px;
    int wc = p & (W_ - 1);

    float ox  = offb[(n << 14) + p];        // x-offset channel n
    float oy  = offb[((n + 9) << 14) + p];  // y-offset channel n+9
    float pxf = ox + (float)(n / 3 - 1) + (float)(h + 1);
    float pyf = oy + (float)(n % 3 - 1) + (float)(wc + 1);

    float pxc = fminf(fmaxf(pxf, 0.f), 129.f);   // Hp-1 = 129
    float pyc = fminf(fmaxf(pyf, 0.f), 129.f);
    float fx = floorf(pxf), fy = floorf(pyf);
    float qx0f = fminf(fmaxf(fx,       0.f), 129.f);
    float qy0f = fminf(fmaxf(fy,       0.f), 129.f);
    float qx1f = fminf(fmaxf(fx + 1.f, 0.f), 129.f);
    float qy1f = fminf(fmaxf(fy + 1.f, 0.f), 129.f);
    int qx0 = (int)qx0f, qy0 = (int)qy0f, qx1 = (int)qx1f, qy1 = (int)qy1f;

    float g_lt = (1.f + (qx0f - pxc)) * (1.f + (qy0f - pyc));
    float g_rb = (1.f - (qx1f - pxc)) * (1.f - (qy1f - pyc));
    float g_lb = (1.f + (qx0f - pxc)) * (1.f - (qy1f - pyc));
    float g_rt = (1.f - (qx1f - pxc)) * (1.f + (qy0f - pyc));

    // zero-padding ring: value is 0 outside [1,128] in padded coords,
    // fold that into the weight so only in-bounds gathers contribute.
    bool vx0 = (qx0 >= 1) && (qx0 <= H_);
    bool vx1 = (qx1 >= 1) && (qx1 <= H_);
    bool vy0 = (qy0 >= 1) && (qy0 <= W_);
    bool vy1 = (qy1 >= 1) && (qy1 <= W_);
    float m00 = (vx0 && vy0) ? g_lt : 0.f;
    float m11 = (vx1 && vy1) ? g_rb : 0.f;
    float m01 = (vx0 && vy1) ? g_lb : 0.f;
    float m10 = (vx1 && vy0) ? g_rt : 0.f;

    int cx0 = min(max(qx0 - 1, 0), H_ - 1);
    int cx1 = min(max(qx1 - 1, 0), H_ - 1);
    int cy0 = min(max(qy0 - 1, 0), W_ - 1);
    int cy1 = min(max(qy1 - 1, 0), W_ - 1);
    int i00 = cx0 * W_ + cy0;
    int i11 = cx1 * W_ + cy1;
    int i01 = cx0 * W_ + cy1;
    int i10 = cx1 * W_ + cy0;

    float* bcol = &Bmat[n * 16 + px];
    const float* xc = xb;
    for (int ci = 0; ci < CIN; ++ci) {
      float v = m00 * xc[i00] + m11 * xc[i11] + m01 * xc[i01] + m10 * xc[i10];
      bcol[ci * 144] = v;  // (ci*9+n)*16+px
      xc += HW_;
    }
  }
  __syncthreads();

  // ---- Phase 2: 144 x V_WMMA_F32_16X16X4_F32 ----------------------------
  v8f acc = {};
  int nl  = lane & 15;            // N column (and A's M row)
  int kh2 = (lane >> 4) << 1;     // lanes 16-31 hold K+2/K+3 halves
  const v2f* Aw = Aswz + ((size_t)(wave * NCHUNK) << 5) + lane;

#pragma unroll 4
  for (int kc = 0; kc < NCHUNK; ++kc) {
    v2f a = Aw[(size_t)kc << 5];              // coalesced global b64
    int krow = (kc << 2) + kh2;
    v2f bb;
    bb.x = Bmat[krow * 16 + nl];              // LDS: half-wave bank split
    bb.y = Bmat[(krow + 1) * 16 + nl];
    acc = __builtin_amdgcn_wmma_f32_16x16x4_f32(
        /*neg_a=*/false, a, /*neg_b=*/false, bb,
        /*c_mod=*/(short)0, acc, /*reuse_a=*/false, /*reuse_b=*/false);
  }

  // ---- Store: C/D layout VGPR r -> M = r + 8*(lane>=16), N = lane&15 ----
  int p = p0 + nl;
  int cobase = (wave << 4) + ((lane >> 4) << 3);
  float* ob = out + ((size_t)b * COUT + cobase) * HW_ + p;
#pragma unroll
  for (int r = 0; r < 8; ++r) {
    ob[0] = acc[r];
    ob += HW_;
  }
}

// ---------------------------------------------------------------------------
extern "C" void kernel_launch(void* const* d_in, const int* in_sizes, int n_in,
                              void* d_out, int out_size, void* d_ws, size_t ws_size,
                              hipStream_t stream) {
  const float* x      = (const float*)d_in[0];  // (4,64,128,128)
  const float* p_w    = (const float*)d_in[1];  // (18,64,3,3)
  const float* p_b    = (const float*)d_in[2];  // (18,)
  const float* conv_w = (const float*)d_in[3];  // (64,64,3,3)
  float* out = (float*)d_out;                   // (4,64,128,128)

  // Workspace layout: [0, 147456) swizzled A fragments, then offsets (4.7 MB)
  v2f*   Aswz = (v2f*)d_ws;
  float* off  = (float*)((char*)d_ws + (size_t)4 * NCHUNK * 32 * sizeof(v2f));

  swizzle_A_kernel<<<dim3(72), dim3(256), 0, stream>>>(conv_w, Aswz);
  offset_conv_kernel<<<dim3(NB * HW_ / 256), dim3(256), 0, stream>>>(x, p_w, p_b, off);
  deform_wmma_kernel<<<dim3(NB * HW_ / 16), dim3(128), 0, stream>>>(x, off, Aswz, out);
}